// Decoder_70781061038951
// MI455X (gfx1250) — compile-verified
//
#include <hip/hip_runtime.h>
#include <math.h>

// CDNA5 wave32 fp32 WMMA: D(16x16,f32) = A(16x4,f32) x B(4x16,f32) + C
typedef __attribute__((ext_vector_type(2))) float v2f;
typedef __attribute__((ext_vector_type(8))) float v8f;

#define KC    64            // polynomial K from reference
#define NPTS  128           // 2*K evaluation points
#define KAUG  132           // augmented K: 65 (Xr) + 65 (Xi) + 2 pad (multiple of 4)
#define NAUG  256           // augmented N: 128 real cols || 128 imag cols
#define ROWS_PER_BLOCK 64   // 8 waves; each wave: 32 rows x 16 output cols

// ---------------------------------------------------------------------------
// Build transposed augmented Vandermonde matrix Bt[n][k], n in [0,256), k in
// [0,132), stored row-major over k (so each WMMA B-fragment = one b64 load).
//   cols n<128  : real-part accumulation at point p=n
//   cols n>=128 : imag-part accumulation at point p=n-128
//   rows k<65   : coefficient from Xr  -> ( Re(w), Im(w) )
//   rows 65..129: coefficient from Xi  -> ( -Im(w), Re(w) )   [i * w]
//   rows >=130  : zero pad
// where w = z_p^(64-k'), z_p = (p<64 ? 1/r : r) * exp(i*2*pi*(p%64)/64),
// r = sqrt(1 + sin(pi/64)).
// ---------------------------------------------------------------------------
__global__ __launch_bounds__(256) void build_vandermonde(float* __restrict__ Bt) {
  int idx = blockIdx.x * 256 + threadIdx.x;
  if (idx >= NAUG * KAUG) return;
  int n  = idx / KAUG;
  int kk = idx - n * KAUG;
  float val = 0.0f;
  if (kk < 2 * KC + 2) {  // kk < 130
    int  kp = (kk < KC + 1) ? kk : (kk - (KC + 1));
    int  e  = KC - kp;                       // exponent 0..64
    int  p  = n & (NPTS - 1);                // point index 0..127
    int  pa = p & (KC - 1);                  // angle index 0..63
    double r   = sqrt(1.0 + sin(M_PI / (double)KC));
    double mag = (p < KC) ? (1.0 / r) : r;
    int    t   = (pa * e) & (KC - 1);        // reduce angle mod 2*pi
    double ang = 2.0 * M_PI * (double)t / (double)KC;
    double me  = pow(mag, (double)e);
    float  c   = (float)(me * cos(ang));
    float  s   = (float)(me * sin(ang));
    bool imagCol = (n >= NPTS);
    bool fromXi  = (kk >= KC + 1);
    val = fromXi ? (imagCol ? c : -s) : (imagCol ? s : c);
  }
  Bt[idx] = val;
}

// ---------------------------------------------------------------------------
// Main GEMM + magnitude-compare kernel.
// Block: 256 threads = 8 waves, covers 64 rows.
//   wave = rg*4 + tg :  rg in {0,1} selects a 32-row group,
//                       tg in {0..3} selects output-column group (16 cols).
// Register blocking: each wave carries TWO 16-row A tiles against the same
// four B fragments -> 8 WMMAs per 4 global b64 loads (2 matrix ops per VMEM).
// Tiles per wave (8 x v8f accumulators):
//   accR0_0/1 = Re at z0-points (cols 16*tg       ), rows lo/hi
//   accR1_0/1 = Re at z1-points (cols 16*tg + 64  )
//   accI0_0/1 = Im at z0-points (cols 16*tg + 128 )
//   accI1_0/1 = Im at z1-points (cols 16*tg + 192 )
// then m = (r^128 * (Re0^2+Im0^2) >= Re1^2+Im1^2) ? 1 : 0.
// ---------------------------------------------------------------------------
__global__ __launch_bounds__(256) void poly_eval_wmma(
    const float* __restrict__ xr, const float* __restrict__ xi,
    const float* __restrict__ Bt, float* __restrict__ out) {
  __shared__ float As[ROWS_PER_BLOCK * KAUG];   // 64*132*4 = 33792 B

  const int tid  = threadIdx.x;
  const int row0 = blockIdx.x * ROWS_PER_BLOCK;

  // Stage augmented A into LDS: As[r][k] = k<65 ? xr[row][k] : k<130 ? xi[row][k-65] : 0
  for (int i = tid; i < ROWS_PER_BLOCK * KAUG; i += 256) {
    int rr = i / KAUG;
    int k  = i - rr * KAUG;
    int gr = row0 + rr;
    float v = 0.0f;
    if (k < KC + 1)            v = xr[(size_t)gr * (KC + 1) + k];
    else if (k < 2 * KC + 2)   v = xi[(size_t)gr * (KC + 1) + (k - (KC + 1))];
    As[i] = v;
  }
  __syncthreads();

  const int wave = tid >> 5;
  const int lane = tid & 31;
  const int rg   = wave >> 2;       // 32-row group 0..1
  const int tg   = wave & 3;        // column group 0..3
  const int half = lane >> 4;       // 0: K=0,1   1: K=2,3  (f32 A/B fragment layout)
  const int li   = lane & 15;       // M (for A) / N (for B) index

  v8f aR0_0 = {}, aR1_0 = {}, aI0_0 = {}, aI1_0 = {};   // rows lo (16)
  v8f aR0_1 = {}, aR1_1 = {}, aI0_1 = {}, aI1_1 = {};   // rows hi (16)

  // A fragment sources: rows (rg*32+li) and (rg*32+16+li), K pair at 2*half
  const float* Arow0 = &As[(rg * 32 + li) * KAUG + 2 * half];
  const float* Arow1 = Arow0 + 16 * KAUG;
  // B fragment sources: column n * 132 + K pair; K-contiguous => single b64
  const int nc = tg * 16 + li;
  const float* B0 = &Bt[(size_t)(nc)       * KAUG + 2 * half];
  const float* B1 = &Bt[(size_t)(nc +  64) * KAUG + 2 * half];
  const float* B2 = &Bt[(size_t)(nc + 128) * KAUG + 2 * half];
  const float* B3 = &Bt[(size_t)(nc + 192) * KAUG + 2 * half];

  for (int k0 = 0; k0 < KAUG; k0 += 4) {
    v2f a0 = *(const v2f*)(Arow0 + k0);    // ds_load_2addr_b64 pair (merged)
    v2f a1 = *(const v2f*)(Arow1 + k0);
    v2f b0 = *(const v2f*)(B0 + k0);       // global_load_b64 (L2-resident)
    v2f b1 = *(const v2f*)(B1 + k0);
    v2f b2 = *(const v2f*)(B2 + k0);
    v2f b3 = *(const v2f*)(B3 + k0);
    aR0_0 = __builtin_amdgcn_wmma_f32_16x16x4_f32(false, a0, false, b0, (short)0, aR0_0, false, false);
    aR0_1 = __builtin_amdgcn_wmma_f32_16x16x4_f32(false, a1, false, b0, (short)0, aR0_1, false, false);
    aR1_0 = __builtin_amdgcn_wmma_f32_16x16x4_f32(false, a0, false, b1, (short)0, aR1_0, false, false);
    aR1_1 = __builtin_amdgcn_wmma_f32_16x16x4_f32(false, a1, false, b1, (short)0, aR1_1, false, false);
    aI0_0 = __builtin_amdgcn_wmma_f32_16x16x4_f32(false, a0, false, b2, (short)0, aI0_0, false, false);
    aI0_1 = __builtin_amdgcn_wmma_f32_16x16x4_f32(false, a1, false, b2, (short)0, aI0_1, false, false);
    aI1_0 = __builtin_amdgcn_wmma_f32_16x16x4_f32(false, a0, false, b3, (short)0, aI1_0, false, false);
    aI1_1 = __builtin_amdgcn_wmma_f32_16x16x4_f32(false, a1, false, b3, (short)0, aI1_1, false, false);
  }

  // scale^2 = r^128 = (1 + sin(pi/64))^64  (compare squared magnitudes)
  const float scale2 = (float)pow(1.0 + sin(M_PI / (double)KC), (double)KC);

  // C/D layout: VGPR v holds row M = v + 8*half, col N = li
  const int colOut = tg * 16 + li;
  for (int v = 0; v < 8; ++v) {
    int growL = row0 + rg * 32 + v + 8 * half;        // lo tile rows
    float R0 = aR0_0[v], R1 = aR1_0[v], I0 = aI0_0[v], I1 = aI1_0[v];
    float m = (scale2 * (R0 * R0 + I0 * I0) >= (R1 * R1 + I1 * I1)) ? 1.0f : 0.0f;
    out[(size_t)growL * KC + colOut] = m;

    int growH = growL + 16;                           // hi tile rows
    float S0 = aR0_1[v], S1 = aR1_1[v], J0 = aI0_1[v], J1 = aI1_1[v];
    float n2 = (scale2 * (S0 * S0 + J0 * J0) >= (S1 * S1 + J1 * J1)) ? 1.0f : 0.0f;
    out[(size_t)growH * KC + colOut] = n2;
  }
}

extern "C" void kernel_launch(void* const* d_in, const int* in_sizes, int n_in,
                              void* d_out, int out_size, void* d_ws, size_t ws_size,
                              hipStream_t stream) {
  const float* xr = (const float*)d_in[0];   // (B, 65) fp32
  const float* xi = (const float*)d_in[1];   // (B, 65) fp32
  float* out = (float*)d_out;                // (B, 64) fp32
  float* Bt  = (float*)d_ws;                 // 256*132 fp32 = 135168 B scratch

  const int Brows = in_sizes[0] / (KC + 1);  // 262144

  const int nB = NAUG * KAUG;
  build_vandermonde<<<(nB + 255) / 256, 256, 0, stream>>>(Bt);
  poly_eval_wmma<<<Brows / ROWS_PER_BLOCK, 256, 0, stream>>>(xr, xi, Bt, out);
}